// SAGEEncoder_37735582662835
// MI455X (gfx1250) — compile-verified
//
#include <hip/hip_runtime.h>
#include <hip/hip_bf16.h>

// ---------------------------------------------------------------------------
// GraphSAGE 2-layer encoder for MI455X (gfx1250), wave32.
// Bandwidth-bound (edge gather/scatter ~2GB >> 3.3 GFLOP of GEMM), so:
//   - edge aggregation via global_atomic_add_f32 into an L2-resident buffer
//     (25.6MB agg buffer << 192MB L2); degree counting fused into pass 1
//   - dense projections via V_WMMA_F32_16X16X4_F32 (exact f32 path, K=64
//     tiled as 16 x (K=4)), fused deg-normalize + ReLU epilogue
//   - per-graph max pool: graph_ids is sorted, so pre-reduce a 64-node strip
//     per thread and flush ~1 atomic per segment (64x fewer atomics)
// ---------------------------------------------------------------------------

typedef __attribute__((ext_vector_type(2))) float v2f;
typedef __attribute__((ext_vector_type(8))) float v8f;

#define NN 100000
#define NE 1600000
#define DF 64
#define NG 64

// ---- helpers: order-preserving float<->uint mapping for atomic max --------
__device__ __forceinline__ unsigned f2mono(float f) {
  unsigned u = __float_as_uint(f);
  return (u & 0x80000000u) ? ~u : (u | 0x80000000u);
}
__device__ __forceinline__ float mono2f(unsigned m) {
  unsigned b = (m & 0x80000000u) ? (m ^ 0x80000000u) : ~m;
  return __uint_as_float(b);
}

// ---- zero / init kernels ---------------------------------------------------
__global__ void k_zero_f(float* __restrict__ p, int n) {
  int i = blockIdx.x * blockDim.x + threadIdx.x;
  if (i < n) p[i] = 0.0f;
}
__global__ void k_init_max(unsigned* __restrict__ p, int n) {
  int i = blockIdx.x * blockDim.x + threadIdx.x;
  if (i < n) p[i] = 0x007FFFFFu;  // f2mono(-inf)
}

// ---- edge scatter: agg[dst] += H[src]; one wave (32 lanes) per edge --------
// Optionally also counts in-degree (deg != nullptr on the first pass).
__global__ void k_scatter(const float* __restrict__ H,
                          const int* __restrict__ src,
                          const int* __restrict__ dst,
                          float* __restrict__ agg,
                          float* __restrict__ deg, int n_edges) {
  const int lane = threadIdx.x & 31;
  const int wid  = (blockIdx.x * blockDim.x + threadIdx.x) >> 5;
  const int nw   = (gridDim.x * blockDim.x) >> 5;
  for (int e = wid; e < n_edges; e += nw) {
    const int s = src[e];
    const int d = dst[e];
    const float v0 = H[(size_t)s * DF + lane];
    const float v1 = H[(size_t)s * DF + lane + 32];
    if (deg && lane == 0) atomicAdd(&deg[d], 1.0f);
    atomicAdd(&agg[(size_t)d * DF + lane], v0);
    atomicAdd(&agg[(size_t)d * DF + lane + 32], v1);
  }
}

// ---- fused SAGE projection: Out = act(Hin@Ws + (Agg/deg)@Wn) ---------------
// One wave per 16-node tile; 8 waves (256 threads) per block.
// A frag (16x4 f32):   lane(m,kh) holds {A[m][k0+2kh], A[m][k0+2kh+1]}
// B frag (4x16 f32):   VGPR0 rows {k0, k0+2} by lane-half; VGPR1 rows {+1,+3}
// C/D  (16x16 f32x8):  VGPR r, half kh -> row r+8kh, col m
__global__ __launch_bounds__(256) void k_sage_gemm(
    const float* __restrict__ Hin, const float* __restrict__ Agg,
    const float* __restrict__ Deg, const float* __restrict__ Wself,
    const float* __restrict__ Wneigh, float* __restrict__ Out, int relu) {
  const int lane = threadIdx.x & 31;
  const int wid  = threadIdx.x >> 5;
  const int tile = blockIdx.x * 8 + wid;
  const int m  = lane & 15;
  const int kh = lane >> 4;

  int row = tile * 16 + m;
  if (row > NN - 1) row = NN - 1;  // clamp loads; stores guarded below

  const float inv = 1.0f / fmaxf(Deg[row], 1.0f);

  // Preload this lane's A fragments for all 16 K-chunks (self + neigh).
  v2f aS[16], aN[16];
#pragma unroll
  for (int kc = 0; kc < 16; ++kc) {
    const float* hp = Hin + (size_t)row * DF + kc * 4 + kh * 2;
    const float* ap = Agg + (size_t)row * DF + kc * 4 + kh * 2;
    v2f s; s.x = hp[0];       s.y = hp[1];
    v2f n; n.x = ap[0] * inv; n.y = ap[1] * inv;
    aS[kc] = s;
    aN[kc] = n;
  }

#pragma unroll
  for (int nt = 0; nt < 4; ++nt) {
    v8f acc = {};
    const int col = nt * 16 + m;
#pragma unroll
    for (int kc = 0; kc < 16; ++kc) {
      const int k0 = kc * 4 + kh * 2;
      v2f bS; bS.x = Wself [(size_t)(k0 + 0) * DF + col];
              bS.y = Wself [(size_t)(k0 + 1) * DF + col];
      v2f bN; bN.x = Wneigh[(size_t)(k0 + 0) * DF + col];
              bN.y = Wneigh[(size_t)(k0 + 1) * DF + col];
      acc = __builtin_amdgcn_wmma_f32_16x16x4_f32(
          false, aS[kc], false, bS, (short)0, acc, false, false);
      acc = __builtin_amdgcn_wmma_f32_16x16x4_f32(
          false, aN[kc], false, bN, (short)0, acc, false, false);
    }
#pragma unroll
    for (int r = 0; r < 8; ++r) {
      const int orow = tile * 16 + r + kh * 8;
      float v = acc[r];
      if (relu) v = fmaxf(v, 0.0f);
      if (orow < NN) Out[(size_t)orow * DF + nt * 16 + m] = v;
    }
  }
}

// ---- per-graph max pooling -------------------------------------------------
// graph_ids is sorted: each thread owns feature f over a 64-node strip and
// flushes one atomic per (segment within strip) instead of one per node.
#define STRIP 64
__global__ void k_segmax(const float* __restrict__ E,
                         const int* __restrict__ gid,
                         unsigned* __restrict__ fm) {
  const int t     = blockIdx.x * blockDim.x + threadIdx.x;
  const int f     = t & 63;
  const int strip = t >> 6;
  const int n0    = strip * STRIP;
  if (n0 >= NN) return;
  const int n1 = (n0 + STRIP < NN) ? (n0 + STRIP) : NN;

  int   curg = gid[n0];
  float cur  = -__builtin_huge_valf();
  for (int n = n0; n < n1; ++n) {
    const int g = gid[n];
    if (g != curg) {
      atomicMax(&fm[curg * DF + f], f2mono(cur));
      curg = g;
      cur  = -__builtin_huge_valf();
    }
    cur = fmaxf(cur, E[(size_t)n * DF + f]);
  }
  atomicMax(&fm[curg * DF + f], f2mono(cur));
}

__global__ void k_decode(const unsigned* __restrict__ fm,
                         float* __restrict__ out) {
  int i = blockIdx.x * blockDim.x + threadIdx.x;
  if (i < NG * DF) out[i] = mono2f(fm[i]);
}

// ---------------------------------------------------------------------------
extern "C" void kernel_launch(void* const* d_in, const int* in_sizes, int n_in,
                              void* d_out, int out_size, void* d_ws,
                              size_t ws_size, hipStream_t stream) {
  const float* x   = (const float*)d_in[0];
  const int*   src = (const int*)d_in[1];
  const int*   dst = (const int*)d_in[2];
  const int*   gid = (const int*)d_in[3];
  const float* W0s = (const float*)d_in[4];
  const float* W0n = (const float*)d_in[5];
  const float* W1s = (const float*)d_in[6];
  const float* W1n = (const float*)d_in[7];

  float* f_out = (float*)d_out;            // [64,64]
  float* e_out = f_out + NG * DF;          // [N,64]

  // workspace layout
  float*    agg = (float*)d_ws;            // N*64
  float*    deg = agg + (size_t)NN * DF;   // N
  float*    h   = deg + NN;                // N*64
  unsigned* fm  = (unsigned*)(h + (size_t)NN * DF);  // 64*64

  const int n_tiles    = (NN + 15) / 16;           // 6250
  const int gemm_grid  = (n_tiles + 7) / 8;        // 782
  const int strips     = (NN + STRIP - 1) / STRIP; // 1563
  const int segmax_thr = strips * DF;

  const int ZN = NN * DF + NN;             // agg + deg are contiguous
  k_zero_f<<<(ZN + 255) / 256, 256, 0, stream>>>(agg, ZN);
  k_init_max<<<(NG * DF + 255) / 256, 256, 0, stream>>>(fm, NG * DF);

  // layer 0 (also accumulates in-degree)
  k_scatter<<<2048, 256, 0, stream>>>(x, src, dst, agg, deg, NE);
  k_sage_gemm<<<gemm_grid, 256, 0, stream>>>(x, agg, deg, W0s, W0n, h,
                                             /*relu=*/1);
  // layer 1
  k_zero_f<<<(NN * DF + 255) / 256, 256, 0, stream>>>(agg, NN * DF);
  k_scatter<<<2048, 256, 0, stream>>>(h, src, dst, agg, nullptr, NE);
  k_sage_gemm<<<gemm_grid, 256, 0, stream>>>(h, agg, deg, W1s, W1n, e_out,
                                             /*relu=*/0);

  // readout
  k_segmax<<<(segmax_thr + 255) / 256, 256, 0, stream>>>(e_out, gid, fm);
  k_decode<<<(NG * DF + 255) / 256, 256, 0, stream>>>(fm, f_out);
}